// Qwen3_5DecoderLayer_85529978732981
// MI455X (gfx1250) — compile-verified
//
#include <hip/hip_runtime.h>

// Gated delta-rule recurrence, exact chunked (WY) formulation for MI455X/gfx1250.
// All dense work runs on V_WMMA_F32_16X16X4_F32 (fp32 tensor path, wave32).
// Global inputs are prefetched one chunk ahead into registers so HBM latency
// hides under the previous chunk's WMMA + forward-substitution work.

typedef float v2f __attribute__((ext_vector_type(2)));
typedef float v4f __attribute__((ext_vector_type(4)));
typedef float v8f __attribute__((ext_vector_type(8)));

namespace {
constexpr int Bb = 4, Ss = 2048, Hh = 16, Dk = 128, Dv = 128;
constexpr int CH  = 16;            // chunk length
constexpr int NCH = Ss / CH;       // 128 chunks
constexpr int SPLIT = 2;           // split Dv across workgroups -> 128 WGs
constexpr int DVL = Dv / SPLIT;    // 64 columns per workgroup
constexpr int NT  = DVL / 16;      // 4 dv tiles

// LDS strides chosen for conflict-free WMMA fragment access patterns
constexpr int SST  = 84;           // state row stride (8*84%64=32, 2*84-16%64=24)
constexpr int QKST = 132;          // q/k row stride  (A-frag column reads conflict-free)
constexpr int DST  = 84;           // [16 x DVL] buffers
constexpr int SMST = 17;           // small 16x16 matrices

constexpr int O_S   = 0;
constexpr int O_Q   = O_S   + Dk * SST;
constexpr int O_K   = O_Q   + CH * QKST;
constexpr int O_V   = O_K   + CH * QKST;
constexpr int O_P   = O_V   + CH * DST;    // K @ S0
constexpr int O_R   = O_P   + CH * DST;    // Q @ S0
constexpr int O_D   = O_R   + CH * DST;    // delta
constexpr int O_DS  = O_D   + CH * DST;    // delta' (state-update scaled)
constexpr int O_AM  = O_DS  + CH * DST;    // K K^T (k-half 1)
constexpr int O_AM2 = O_AM  + CH * SMST;   // K K^T (k-half 2)
constexpr int O_BM  = O_AM2 + CH * SMST;   // Q K^T (k-half 1)
constexpr int O_BM2 = O_BM  + CH * SMST;   // Q K^T (k-half 2)
constexpr int O_MS  = O_BM2 + CH * SMST;   // output mixing matrix (masked, scaled)
constexpr int O_LM  = O_MS  + CH * SMST;   // forward-substitution matrix (strict lower)
constexpr int O_RQ  = O_LM  + CH * SMST;   // 16x16 sumsq partials (q)
constexpr int O_RK  = O_RQ  + 256;         // 16x16 sumsq partials (k)
constexpr int O_IQ  = O_RK  + 256;         // rsqrt(|q|^2+eps)*scale
constexpr int O_IK  = O_IQ  + 16;          // rsqrt(|k|^2+eps)
constexpr int O_AV  = O_IK  + 16;          // exp(g)
constexpr int O_AVI = O_AV  + 16;          // exp(-g)
constexpr int O_BT  = O_AVI + 16;          // beta
constexpr int O_OQ  = O_BT  + 16;          // c[t]*invq[t] for output diag term
constexpr int O_CC  = O_OQ  + 16;          // c[15]
constexpr int LDS_FLOATS = O_CC + 4;
} // namespace

__device__ __forceinline__ v8f zero8() {
  v8f z;
#pragma unroll
  for (int i = 0; i < 8; ++i) z[i] = 0.f;
  return z;
}

// A-fragment (16x4, MxK): lane l, vgpr r -> A[l%16][kb + 2*(l>>4) + r]
__device__ __forceinline__ v2f ldsA(const float* p, int stride, int m0, int kb, int lane) {
  const float* q = p + (m0 + (lane & 15)) * stride + kb + 2 * (lane >> 4);
  v2f r; r.x = q[0]; r.y = q[1]; return r;
}
// B-fragment (4x16, KxN): lane l, vgpr r -> B[kb + 2*(l>>4) + r][l%16]
__device__ __forceinline__ v2f ldsB(const float* p, int stride, int kb, int n0, int lane) {
  const float* q = p + (kb + 2 * (lane >> 4)) * stride + n0 + (lane & 15);
  v2f r; r.x = q[0]; r.y = q[stride]; return r;
}
// C/D fragment (16x16): lane l, vgpr r -> C[r + 8*(l>>4)][l%16]
__device__ __forceinline__ v8f ldsC(const float* p, int stride, int m0, int n0, int lane) {
  const int mb = m0 + 8 * (lane >> 4), n = n0 + (lane & 15);
  v8f r;
#pragma unroll
  for (int i = 0; i < 8; ++i) r[i] = p[(mb + i) * stride + n];
  return r;
}
__device__ __forceinline__ void stsC(float* p, int stride, int m0, int n0, int lane, v8f v) {
  const int mb = m0 + 8 * (lane >> 4), n = n0 + (lane & 15);
#pragma unroll
  for (int i = 0; i < 8; ++i) p[(mb + i) * stride + n] = v[i];
}

__device__ __forceinline__ v8f wmma4(v2f a, v2f b, v8f c) {
  // D = A(16x4,f32) * B(4x16,f32) + C(16x16,f32)
  return __builtin_amdgcn_wmma_f32_16x16x4_f32(false, a, false, b, (short)0, c,
                                               false, false);
}

__global__ void __launch_bounds__(256, 1)
gdn_chunk_kernel(const float* __restrict__ qg, const float* __restrict__ kg,
                 const float* __restrict__ vg, const float* __restrict__ gg,
                 const float* __restrict__ bg, float* __restrict__ out) {
  extern __shared__ float sm[];
  const int tid  = threadIdx.x;
  const int lane = tid & 31;
  const int w    = tid >> 5;

  const int wg    = blockIdx.x;
  const int half  = wg & (SPLIT - 1);
  const int bh    = wg >> 1;            // SPLIT == 2
  const int b     = bh >> 4;            // / Hh
  const int h     = bh & 15;            // % Hh
  const int dvoff = half * DVL;

  float* Ssm = sm + O_S;
  float* Qs  = sm + O_Q;
  float* Ks  = sm + O_K;

  const int t = tid >> 4;               // chunk row handled by this thread
  const int p = tid & 15;

  constexpr float SCALE = 0.08838834764831845f; // 1/sqrt(128)

  // ---- prefetch chunk 0 into registers ----
  v4f pq0, pq1, pk0, pk1, pv0;
  float pg = 0.f, pb = 0.f;
  {
    const size_t rowbase = (size_t)((b * Ss + t) * Hh + h);
    const float* qrow = qg + rowbase * Dk + p * 8;
    const float* krow = kg + rowbase * Dk + p * 8;
    pq0 = *(const v4f*)qrow; pq1 = *(const v4f*)(qrow + 4);
    pk0 = *(const v4f*)krow; pk1 = *(const v4f*)(krow + 4);
    pv0 = *(const v4f*)(vg + rowbase * Dv + dvoff + p * 4);
    if (tid < 16) {
      const size_t gi = (size_t)(b * Ss + tid) * Hh + h;
      pg = gg[gi]; pb = bg[gi];
    }
  }

  // zero initial state (overlaps with prefetch latency)
  for (int i = tid; i < Dk * SST; i += 256) Ssm[i] = 0.f;

  for (int ch = 0; ch < NCH; ++ch) {
    const int t0 = ch * CH;

    // ---- commit prefetched registers to LDS, fused sum-of-squares partials ----
    {
      float sq = 0.f, sk = 0.f;
#pragma unroll
      for (int e = 0; e < 4; ++e) {
        sq += pq0[e] * pq0[e] + pq1[e] * pq1[e];
        sk += pk0[e] * pk0[e] + pk1[e] * pk1[e];
      }
      float* qd = Qs + t * QKST + p * 8;
      float* kd = Ks + t * QKST + p * 8;
      *(v4f*)qd = pq0; *(v4f*)(qd + 4) = pq1;
      *(v4f*)kd = pk0; *(v4f*)(kd + 4) = pk1;
      sm[O_RQ + t * 16 + p] = sq;
      sm[O_RK + t * 16 + p] = sk;
      *(v4f*)(sm + O_V + t * DST + p * 4) = pv0;
      if (tid < 16) {
        sm[O_AV + tid]  = expf(pg);
        sm[O_AVI + tid] = expf(-pg);
        sm[O_BT + tid]  = pb;
      }
    }

    // ---- issue next chunk's global loads (land during this chunk's compute) ----
    {
      const int t0n = (ch + 1 < NCH) ? (t0 + CH) : 0;   // clamped, always in-bounds
      const size_t rowbase = (size_t)((b * Ss + t0n + t) * Hh + h);
      const float* qrow = qg + rowbase * Dk + p * 8;
      const float* krow = kg + rowbase * Dk + p * 8;
      pq0 = *(const v4f*)qrow; pq1 = *(const v4f*)(qrow + 4);
      pk0 = *(const v4f*)krow; pk1 = *(const v4f*)(krow + 4);
      pv0 = *(const v4f*)(vg + rowbase * Dv + dvoff + p * 4);
      if (tid < 16) {
        const size_t gi = (size_t)(b * Ss + t0n + tid) * Hh + h;
        pg = gg[gi]; pb = bg[gi];
      }
    }
    __syncthreads();

    // ---- finalize norms ----
    if (tid < 16) {
      float sq = 0.f, sk = 0.f;
#pragma unroll
      for (int pp = 0; pp < 16; ++pp) {
        sq += sm[O_RQ + tid * 16 + pp];
        sk += sm[O_RK + tid * 16 + pp];
      }
      sm[O_IQ + tid] = rsqrtf(sq + 1e-6f) * SCALE;
      sm[O_IK + tid] = rsqrtf(sk + 1e-6f);
    }
    __syncthreads();

    // ---- WMMA phase 1: P = K@S0, R = Q@S0, KK^T, QK^T (raw, scales folded later)
    if (w < 4) {
      const int n0 = w * 16;
      v8f acc = zero8();
#pragma unroll 8
      for (int kb = 0; kb < Dk; kb += 4)
        acc = wmma4(ldsA(Ks, QKST, 0, kb, lane), ldsB(Ssm, SST, kb, n0, lane), acc);
      stsC(sm + O_P, DST, 0, n0, lane, acc);

      // second job: KK^T / QK^T, each split in k-halves across waves 0..3
      const float* asrc = (w < 2) ? Ks : Qs;   // w0,w1: KK^T ; w2,w3: QK^T
      const int ks = (w & 1) ? 64 : 0;
      v8f acc2 = zero8();
#pragma unroll 8
      for (int kb = ks; kb < ks + 64; kb += 4) {
        v2f a  = ldsA(asrc, QKST, 0, kb, lane);
        v2f bt = ldsA(Ks,   QKST, 0, kb, lane); // B-layout of K^T == A-layout of K
        acc2 = wmma4(a, bt, acc2);
      }
      float* dst = sm + ((w == 0) ? O_AM : (w == 1) ? O_AM2 : (w == 2) ? O_BM : O_BM2);
      stsC(dst, SMST, 0, 0, lane, acc2);
    } else {
      const int n0 = (w - 4) * 16;
      v8f acc = zero8();
#pragma unroll 8
      for (int kb = 0; kb < Dk; kb += 4)
        acc = wmma4(ldsA(Qs, QKST, 0, kb, lane), ldsB(Ssm, SST, kb, n0, lane), acc);
      stsC(sm + O_R, DST, 0, n0, lane, acc);
    }
    __syncthreads();

    // ---- build mixing matrices (norm & cumulative-decay scales folded in) ----
    {
      const int tt = tid >> 4, ii = tid & 15;
      float cc = 1.f, cci = 1.f, c_t = 1.f, ci_i = 1.f, c_i = 1.f;
#pragma unroll
      for (int m = 0; m < 16; ++m) {
        cc  *= sm[O_AV + m];
        cci *= sm[O_AVI + m];
        if (m == tt) c_t = cc;
        if (m == ii) { ci_i = cci; c_i = cc; }
      }
      const float ikt = sm[O_IK + tt], iki = sm[O_IK + ii], iqt = sm[O_IQ + tt];
      const float am = sm[O_AM + tt * SMST + ii] + sm[O_AM2 + tt * SMST + ii];
      const float bm = sm[O_BM + tt * SMST + ii] + sm[O_BM2 + tt * SMST + ii];
      sm[O_LM + tt * SMST + ii] = (ii < tt)  ? c_t * ci_i * ikt * iki * am : 0.f;
      sm[O_MS + tt * SMST + ii] = (ii <= tt) ? c_t * ci_i * iqt * iki * bm : 0.f;
      if (tid < 16) sm[O_OQ + tid] = c_i * sm[O_IQ + tid];
      if (tid == 15) sm[O_CC] = c_i;          // c[15] = chunk-total decay
    }
    __syncthreads();

    // ---- serial core: 16-step forward substitution for delta (per dv column) ----
    if (tid < DVL) {
      const int dv = tid;
      float cv[16], civ[16];
      float cc = 1.f, cci = 1.f;
#pragma unroll
      for (int m = 0; m < 16; ++m) {
        cc *= sm[O_AV + m]; cci *= sm[O_AVI + m];
        cv[m] = cc; civ[m] = cci;
      }
      const float cc15 = cv[15];
      float d[16];
#pragma unroll
      for (int t2 = 0; t2 < 16; ++t2) {
        float kv = cv[t2] * sm[O_IK + t2] * sm[O_P + t2 * DST + dv];
#pragma unroll
        for (int j = 0; j < t2; ++j) kv += sm[O_LM + t2 * SMST + j] * d[j];
        const float dt = sm[O_BT + t2] * (sm[O_V + t2 * DST + dv] - kv);
        d[t2] = dt;
        sm[O_D  + t2 * DST + dv] = dt;
        sm[O_DS + t2 * DST + dv] = cc15 * civ[t2] * sm[O_IK + t2] * dt;
      }
    }
    __syncthreads();

    // ---- WMMA phase 2: outputs + state update ----
    if (w < 4) {  // out = diag(c*invq)*R + M @ Delta, stored straight to global
      const int n0 = w * 16;
      v8f acc = zero8();
#pragma unroll
      for (int kb = 0; kb < 16; kb += 4)
        acc = wmma4(ldsA(sm + O_MS, SMST, 0, kb, lane), ldsB(sm + O_D, DST, kb, n0, lane), acc);
      const int n  = n0 + (lane & 15);
      const int tb = 8 * (lane >> 4);
#pragma unroll
      for (int r = 0; r < 8; ++r) {
        const int tt = tb + r;
        const float val = acc[r] + sm[O_OQ + tt] * sm[O_R + tt * DST + n];
        out[((size_t)((b * Ss + t0 + tt) * Hh + h)) * Dv + dvoff + n] = val;
      }
    }
    {   // S <- c15 * S + K^T @ Delta'   (each wave owns one 16-row dk tile)
      const float cc15 = sm[O_CC];
      const int m0 = w * 16;
      v2f ka[4];   // K^T A-fragments: identical for all 4 dv tiles -> hoist
#pragma unroll
      for (int kk = 0; kk < 4; ++kk) ka[kk] = ldsB(Ks, QKST, kk * 4, m0, lane);
#pragma unroll
      for (int nt = 0; nt < NT; ++nt) {
        const int n0 = nt * 16;
        v8f acc = ldsC(Ssm, SST, m0, n0, lane);
#pragma unroll
        for (int i = 0; i < 8; ++i) acc[i] *= cc15;
#pragma unroll
        for (int kk = 0; kk < 4; ++kk)
          acc = wmma4(ka[kk], ldsB(sm + O_DS, DST, kk * 4, n0, lane), acc);
        stsC(Ssm, SST, m0, n0, lane, acc);
      }
    }
    __syncthreads();
  }

  // ---- final state: [B,H,Dk,Dv] appended after outputs ----
  float* stateout = out + (size_t)Bb * Ss * Hh * Dv;
  for (int i = tid; i < Dk * DVL; i += 256) {
    const int dk = i >> 6, dv = i & (DVL - 1);
    stateout[((size_t)(b * Hh + h) * Dk + dk) * Dv + dvoff + dv] = Ssm[dk * SST + dv];
  }
}

extern "C" void kernel_launch(void* const* d_in, const int* in_sizes, int n_in,
                              void* d_out, int out_size, void* d_ws, size_t ws_size,
                              hipStream_t stream) {
  (void)in_sizes; (void)n_in; (void)out_size; (void)d_ws; (void)ws_size;
  const float* q  = (const float*)d_in[0];
  const float* k  = (const float*)d_in[1];
  const float* v  = (const float*)d_in[2];
  const float* g  = (const float*)d_in[3];
  const float* bt = (const float*)d_in[4];
  float* out = (float*)d_out;

  const int ldsBytes = LDS_FLOATS * (int)sizeof(float);   // ~94 KB < 160 KB
  (void)hipFuncSetAttribute((const void*)gdn_chunk_kernel,
                            hipFuncAttributeMaxDynamicSharedMemorySize, ldsBytes);
  dim3 grid(Bb * Hh * SPLIT);   // 128 workgroups
  dim3 block(256);              // 8 wave32s
  gdn_chunk_kernel<<<grid, block, ldsBytes, stream>>>(q, k, v, g, bt, out);
}